// Interpolate_28664611734214
// MI455X (gfx1250) — compile-verified
//
#include <hip/hip_runtime.h>

// Interpolate: out[j] = sum_n wgt[n] * data[s_n, t_n, u_{n,k}, v_{n,k}, c],  j = k*3+c
// Weighted sum done with native CDNA5 V_WMMA_F32_16X16X4_F32 (K=4 f32 matrix op):
//   A(16x4) = gathered data for 16 consecutive flat outputs, B(4x16) = weights broadcast.

typedef __attribute__((ext_vector_type(2))) float v2f;
typedef __attribute__((ext_vector_type(8))) float v8f;

#define S_DIM 9
#define T_DIM 9
#define U_DIM 512
#define V_DIM 512
#define H_DIM 512
#define W_DIM 512
#define N_NBR 4
#define EPS_F 1e-6f

#define HW      (H_DIM * W_DIM)       // 262144 pixels
#define J_TOT   (HW * 3)              // 786432 flat outputs
#define TILE    16                    // outputs per WMMA
#define N_TILES (J_TOT / TILE)        // 49152
#define TPW     4                     // tiles per wave
#define WPB     8                     // waves per block (wave32)
#define BLOCK   (WPB * 32)            // 256 threads
#define N_BLOCKS (N_TILES / (TPW * WPB))  // 1536

__global__ __launch_bounds__(BLOCK)
void interp_wmma_kernel(const float* __restrict__ data,
                        const float* __restrict__ cam,
                        const int*   __restrict__ nbrs,
                        const int*   __restrict__ uv,
                        float*       __restrict__ out)
{
    // ---- per-wave (redundant, tiny) weight computation -------------------
    const float c0 = cam[0];
    const float c1 = cam[1];

    float q[N_NBR];
    int   sbase[N_NBR];
#pragma unroll
    for (int i = 0; i < N_NBR; ++i) {
        const int s  = nbrs[2 * i + 0];
        const int tt = nbrs[2 * i + 1];
        const float d0 = c0 - (float)s;
        const float d1 = c1 - (float)tt;
        q[i]     = fabsf(d0 * d1);
        sbase[i] = (s * T_DIM + tt) * (U_DIM * V_DIM * 3);
    }
    // flip, epsilon mask, normalize, epsilon mask again (matches reference)
    float tv[N_NBR];
#pragma unroll
    for (int i = 0; i < N_NBR; ++i) {
        const float v = q[N_NBR - 1 - i];
        tv[i] = (v <= EPS_F) ? 0.0f : v;
    }
    const float tsum = tv[0] + tv[1] + tv[2] + tv[3];
    float wgt[N_NBR];
#pragma unroll
    for (int i = 0; i < N_NBR; ++i) {
        const float w = tv[i] / tsum;
        wgt[i] = (fabsf(w) <= EPS_F) ? 0.0f : w;
    }

    // ---- lane roles (wave32) ---------------------------------------------
    const unsigned lane = threadIdx.x & 31u;
    const unsigned wave = threadIdx.x >> 5;
    const int col = (int)(lane & 15u);        // output column within 16-wide tile
    const int hi  = (int)((lane >> 4) & 1u);  // 0: neighbors {0,1}, 1: neighbors {2,3}
    const int n0  = hi * 2;
    const int n1  = n0 + 1;

    // B(4x16): row k broadcast with wgt[k].
    // Layout: V0 = K0 (lanes 0-15) / K2 (lanes 16-31), V1 = K1 / K3.
    v2f b;
    b.x = hi ? wgt[2] : wgt[0];
    b.y = hi ? wgt[3] : wgt[1];

    const int tile0 = ((int)blockIdx.x * WPB + (int)wave) * TPW;
    const int2* __restrict__ uvp = (const int2*)uv;

    // ---- gather A fragments: 2 random 4B loads per lane per tile ---------
    v2f a[TPW];
#pragma unroll
    for (int t = 0; t < TPW; ++t) {
        const int j = (tile0 + t) * TILE + col;   // flat output index
        const int k = j / 3;                      // pixel
        const int c = j - 3 * k;                  // channel
        const int2 uv0 = uvp[n0 * HW + k];        // (u, v) for neighbor n0
        const int2 uv1 = uvp[n1 * HW + k];
        a[t].x = data[sbase[n0] + (uv0.x * V_DIM + uv0.y) * 3 + c];
        a[t].y = data[sbase[n1] + (uv1.x * V_DIM + uv1.y) * 3 + c];
    }

    // ---- K=4 f32 WMMA: D[m,n] = sum_k g[k][j0+m] * wgt[k] ---------------
    // (wave-uniform control flow up to here => EXEC all ones, as required)
    v8f d[TPW];
#pragma unroll
    for (int t = 0; t < TPW; ++t) {
        v8f cz = {};
        d[t] = __builtin_amdgcn_wmma_f32_16x16x4_f32(
            /*neg_a=*/false, a[t],
            /*neg_b=*/false, b,
            /*c_mod=*/(short)0, cz,
            /*reuse_a=*/false, /*reuse_b=*/false);
    }

    // ---- store: D columns are replicated; lane 0 holds M=0..7, lane 16 M=8..15
    if (col == 0) {
#pragma unroll
        for (int t = 0; t < TPW; ++t) {
            float* op = out + (tile0 + t) * TILE + hi * 8;
            float4 p0 = make_float4(d[t][0], d[t][1], d[t][2], d[t][3]);
            float4 p1 = make_float4(d[t][4], d[t][5], d[t][6], d[t][7]);
            *(float4*)(op + 0) = p0;
            *(float4*)(op + 4) = p1;
        }
    }
}

extern "C" void kernel_launch(void* const* d_in, const int* in_sizes, int n_in,
                              void* d_out, int out_size, void* d_ws, size_t ws_size,
                              hipStream_t stream) {
    const float* data = (const float*)d_in[0];  // (9,9,512,512,3) f32
    // d_in[1] = pixel (512,512) f32 — only its shape is used by the reference
    const float* cam  = (const float*)d_in[2];  // (3,) f32
    const int*   nbr  = (const int*)d_in[3];    // (4,2) i32
    const int*   uv   = (const int*)d_in[4];    // (4,512,512,2) i32
    float*       out  = (float*)d_out;          // 786432 f32 (viewed as (3,512,512))

    (void)in_sizes; (void)n_in; (void)out_size; (void)d_ws; (void)ws_size;

    interp_wmma_kernel<<<N_BLOCKS, BLOCK, 0, stream>>>(data, cam, nbr, uv, out);
}